// GatedGCNNet_79877801771059
// MI455X (gfx1250) — compile-verified
//
#include <hip/hip_runtime.h>
#include <hip/hip_bf16.h>
#include <math.h>

// ---------------------------------------------------------------------------
// GatedGCN (N=50000, E=800000, D=64, L=4) on gfx1250.
// Dense GEMMs via V_WMMA_F32_16X16X4_F32; edge gather/scatter via global
// f32 atomics; batchnorm stats via LDS ds_add_f32 + global atomics.
// v2: phase-split node GEMM + __launch_bounds__ to kill VGPR spills.
// ---------------------------------------------------------------------------

#define DD 64
#define KPAD 66            // k-dim pad for LDS weight slabs (bank-conflict avoidance)
#define SLAB (DD * KPAD)   // 4224 floats per 64x64 weight slab

typedef __attribute__((ext_vector_type(2))) float v2f;
typedef __attribute__((ext_vector_type(8))) float v8f;

// Stage a 64x64 row-major weight W[k][c] into LDS as ls[c*KPAD + k]
__device__ __forceinline__ void stage_w(const float* __restrict__ g, float* ls,
                                        int tid, int nthreads) {
  for (int i = tid; i < DD * DD; i += nthreads) {
    int k = i >> 6, c = i & 63;
    ls[c * KPAD + k] = g[i];
  }
}

// One 16x64 @ 64x64 (+bias) tile, D-matrix stored to out (row-major [*,64]).
// af: 16 A-fragments (v2f each) for this wave's 16-row tile.
__device__ __forceinline__ void gemm_tile(const v2f* af, const float* lw,
                                          const float* __restrict__ bias,
                                          float* __restrict__ out,
                                          int row0, int m, int hi) {
#pragma unroll
  for (int nn = 0; nn < 4; ++nn) {
    int col = nn * 16 + m;
    float bv = bias[col];
    v8f acc = {bv, bv, bv, bv, bv, bv, bv, bv};
    const float* wc = lw + col * KPAD + 2 * hi;
#pragma unroll
    for (int kk = 0; kk < 16; ++kk) {
      v2f b = *(const v2f*)(wc + 4 * kk);
      acc = __builtin_amdgcn_wmma_f32_16x16x4_f32(false, af[kk], false, b,
                                                  (short)0, acc, false, false);
    }
#pragma unroll
    for (int r = 0; r < 8; ++r)
      out[(row0 + hi * 8 + r) * DD + col] = acc[r];
  }
}

// Accumulating variant: out-tile is loaded as the WMMA C operand, then
// out += af @ W (used for the p-half of the A2 concat GEMM).
__device__ __forceinline__ void gemm_tile_acc(const v2f* af, const float* lw,
                                              float* __restrict__ out,
                                              int row0, int m, int hi) {
#pragma unroll
  for (int nn = 0; nn < 4; ++nn) {
    int col = nn * 16 + m;
    v8f acc;
#pragma unroll
    for (int r = 0; r < 8; ++r)
      acc[r] = out[(row0 + hi * 8 + r) * DD + col];
    const float* wc = lw + col * KPAD + 2 * hi;
#pragma unroll
    for (int kk = 0; kk < 16; ++kk) {
      v2f b = *(const v2f*)(wc + 4 * kk);
      acc = __builtin_amdgcn_wmma_f32_16x16x4_f32(false, af[kk], false, b,
                                                  (short)0, acc, false, false);
    }
#pragma unroll
    for (int r = 0; r < 8; ++r)
      out[(row0 + hi * 8 + r) * DD + col] = acc[r];
  }
}

// --------------------------- kernels ---------------------------------------

__global__ void k_zero(float* __restrict__ x, int n) {
  for (int i = blockIdx.x * blockDim.x + threadIdx.x; i < n;
       i += gridDim.x * blockDim.x)
    x[i] = 0.0f;
}

__global__ void k_embed_nodes(const int* __restrict__ h_ids,
                              const int* __restrict__ f_ids,
                              const float* __restrict__ emb_h,
                              const float* __restrict__ emb_f,
                              float* __restrict__ h, float* __restrict__ p,
                              int total) {
  for (int i = blockIdx.x * blockDim.x + threadIdx.x; i < total;
       i += gridDim.x * blockDim.x) {
    int n = i >> 6, c = i & 63;
    h[i] = emb_h[h_ids[n] * DD + c];
    p[i] = emb_f[f_ids[n] * DD + c];
  }
}

__global__ void k_embed_edges(const int* __restrict__ e_ids,
                              const float* __restrict__ emb_e,
                              float* __restrict__ e, int total) {
  for (int i = blockIdx.x * blockDim.x + threadIdx.x; i < total;
       i += gridDim.x * blockDim.x)
    e[i] = emb_e[e_ids[i >> 6] * DD + (i & 63)];
}

// All six node-side projections in one pass over h,p.
// Phase 1 uses h-fragments (B1, B2, A1, A2-top); phase 2 reuses the same
// fragment registers for p (A2-bottom accumulate, C1, C2).
__global__ void __launch_bounds__(256, 1)
k_node_gemms(const float* __restrict__ h, const float* __restrict__ p,
             const float* __restrict__ B1w, const float* __restrict__ B1b,
             const float* __restrict__ B2w, const float* __restrict__ B2b,
             const float* __restrict__ A1w, const float* __restrict__ A1b,
             const float* __restrict__ A2w, const float* __restrict__ A2b,
             const float* __restrict__ C1w, const float* __restrict__ C1b,
             const float* __restrict__ C2w, const float* __restrict__ C2b,
             float* __restrict__ B1h, float* __restrict__ B2h,
             float* __restrict__ h_new, float* __restrict__ vbuf,
             float* __restrict__ p_new, float* __restrict__ C2p, int ntiles) {
  extern __shared__ float sm[];
  float* sB1 = sm;
  float* sB2 = sB1 + SLAB;
  float* sA1 = sB2 + SLAB;
  float* sA2t = sA1 + SLAB;
  float* sA2b = sA2t + SLAB;
  float* sC1 = sA2b + SLAB;
  float* sC2 = sC1 + SLAB;
  int tid = threadIdx.x, nth = blockDim.x;
  stage_w(B1w, sB1, tid, nth);
  stage_w(B2w, sB2, tid, nth);
  stage_w(A1w, sA1, tid, nth);
  stage_w(A2w, sA2t, tid, nth);             // rows 0..63 (h half)
  stage_w(A2w + DD * DD, sA2b, tid, nth);   // rows 64..127 (p half)
  stage_w(C1w, sC1, tid, nth);
  stage_w(C2w, sC2, tid, nth);
  __syncthreads();

  int lane = tid & 31, w = tid >> 5;
  int m = lane & 15, hi = lane >> 4;
  int wpb = blockDim.x >> 5;
  for (int tile = blockIdx.x * wpb + w; tile < ntiles; tile += gridDim.x * wpb) {
    int row0 = tile * 16;
    v2f af[16];
    // ---- phase 1: h fragments ----
    {
      const float* hrow = h + (row0 + m) * DD + 2 * hi;
#pragma unroll
      for (int kk = 0; kk < 16; ++kk) af[kk] = *(const v2f*)(hrow + 4 * kk);
    }
    gemm_tile(af, sB1, B1b, B1h, row0, m, hi);
    gemm_tile(af, sB2, B2b, B2h, row0, m, hi);
    gemm_tile(af, sA1, A1b, h_new, row0, m, hi);   // h_new = A1h + b
    gemm_tile(af, sA2t, A2b, vbuf, row0, m, hi);   // v  = h@A2_top + b
    // ---- phase 2: p fragments (reuse af registers) ----
    {
      const float* prow = p + (row0 + m) * DD + 2 * hi;
#pragma unroll
      for (int kk = 0; kk < 16; ++kk) af[kk] = *(const v2f*)(prow + 4 * kk);
    }
    gemm_tile_acc(af, sA2b, vbuf, row0, m, hi);    // v += p@A2_bot
    gemm_tile(af, sC1, C1b, p_new, row0, m, hi);   // p_new = C1p + b
    gemm_tile(af, sC2, C2b, C2p, row0, m, hi);
  }
}

// hat_eta = e@B3 + b + B1h[src] + B2h[dst]; sigma scatter-summed at dst.
__global__ void __launch_bounds__(256, 1)
k_edge_gate(const float* __restrict__ e, const float* __restrict__ B3w,
            const float* __restrict__ B3b, const float* __restrict__ B1h,
            const float* __restrict__ B2h, const int* __restrict__ src,
            const int* __restrict__ dst, float* __restrict__ hat,
            float* __restrict__ sum_sigma, int ntiles) {
  __shared__ float sB3[SLAB];
  stage_w(B3w, sB3, threadIdx.x, blockDim.x);
  __syncthreads();
  int lane = threadIdx.x & 31, w = threadIdx.x >> 5;
  int m = lane & 15, hi = lane >> 4;
  int wpb = blockDim.x >> 5;
  for (int tile = blockIdx.x * wpb + w; tile < ntiles; tile += gridDim.x * wpb) {
    int row0 = tile * 16;
    v2f ae[16];
    const float* erow = e + (row0 + m) * DD + 2 * hi;
#pragma unroll
    for (int kk = 0; kk < 16; ++kk) ae[kk] = *(const v2f*)(erow + 4 * kk);
#pragma unroll
    for (int nn = 0; nn < 4; ++nn) {
      int col = nn * 16 + m;
      float bv = B3b[col];
      v8f acc = {bv, bv, bv, bv, bv, bv, bv, bv};
      const float* wc = sB3 + col * KPAD + 2 * hi;
#pragma unroll
      for (int kk = 0; kk < 16; ++kk) {
        v2f b = *(const v2f*)(wc + 4 * kk);
        acc = __builtin_amdgcn_wmma_f32_16x16x4_f32(false, ae[kk], false, b,
                                                    (short)0, acc, false, false);
      }
#pragma unroll
      for (int r = 0; r < 8; ++r) {
        int row = row0 + hi * 8 + r;
        int s = src[row], d = dst[row];
        float he = acc[r] + B1h[s * DD + col] + B2h[d * DD + col];
        hat[row * DD + col] = he;
        float sg = 1.0f / (1.0f + expf(-he));
        atomicAdd(&sum_sigma[d * DD + col], sg);
      }
    }
  }
}

// eta = sigma/(sum_sigma[dst]+eps); scatter eta*v[src] -> h_new, eta*C2p[src] -> p_new
__global__ void k_edge_scatter(const float* __restrict__ hat,
                               const float* __restrict__ sum_sigma,
                               const float* __restrict__ vbuf,
                               const float* __restrict__ C2p,
                               const int* __restrict__ src,
                               const int* __restrict__ dst,
                               float* __restrict__ h_new,
                               float* __restrict__ p_new, int total) {
  for (int i = blockIdx.x * blockDim.x + threadIdx.x; i < total;
       i += gridDim.x * blockDim.x) {
    int ed = i >> 6, c = i & 63;
    float he = hat[i];
    float sg = 1.0f / (1.0f + expf(-he));
    int s = src[ed], d = dst[ed];
    float eta = sg / (sum_sigma[d * DD + c] + 1e-6f);
    atomicAdd(&h_new[d * DD + c], eta * vbuf[s * DD + c]);
    atomicAdd(&p_new[d * DD + c], eta * C2p[s * DD + c]);
  }
}

// Per-column sum & sumsq (for batchnorm over axis 0). blockDim must be a
// multiple of 64 so each thread's column stays fixed across the stride loop.
__global__ void k_colstats(const float* __restrict__ x, int total,
                           float* __restrict__ out /* [0:64)=sum [64:128)=sq */) {
  __shared__ float ls[128];
  if (threadIdx.x < 128) ls[threadIdx.x] = 0.0f;
  __syncthreads();
  int i0 = blockIdx.x * blockDim.x + threadIdx.x;
  int c = i0 & 63;
  float s = 0.0f, q = 0.0f;
  for (int i = i0; i < total; i += gridDim.x * blockDim.x) {
    float v = x[i];
    s += v;
    q += v * v;
  }
  atomicAdd(&ls[c], s);
  atomicAdd(&ls[64 + c], q);
  __syncthreads();
  if (threadIdx.x < 128) atomicAdd(&out[threadIdx.x], ls[threadIdx.x]);
}

// h += relu(bn(h_new)); p += tanh(p_new)
__global__ void k_finalize_node(float* __restrict__ h, float* __restrict__ p,
                                const float* __restrict__ h_new,
                                const float* __restrict__ p_new,
                                const float* __restrict__ stats,
                                const float* __restrict__ g,
                                const float* __restrict__ b, float invn,
                                int total) {
  for (int i = blockIdx.x * blockDim.x + threadIdx.x; i < total;
       i += gridDim.x * blockDim.x) {
    int c = i & 63;
    float mu = stats[c] * invn;
    float var = stats[64 + c] * invn - mu * mu;
    float t = (h_new[i] - mu) * rsqrtf(var + 1e-5f) * g[c] + b[c];
    h[i] += fmaxf(t, 0.0f);
    p[i] += tanhf(p_new[i]);
  }
}

// e += relu(bn(hat_eta))
__global__ void k_finalize_edge(float* __restrict__ e,
                                const float* __restrict__ hat,
                                const float* __restrict__ stats,
                                const float* __restrict__ g,
                                const float* __restrict__ b, float invn,
                                int total) {
  for (int i = blockIdx.x * blockDim.x + threadIdx.x; i < total;
       i += gridDim.x * blockDim.x) {
    int c = i & 63;
    float mu = stats[c] * invn;
    float var = stats[64 + c] * invn - mu * mu;
    float t = (hat[i] - mu) * rsqrtf(var + 1e-5f) * g[c] + b[c];
    e[i] += fmaxf(t, 0.0f);
  }
}

// mean readout + 64->32->16->1 MLP (single block of 64 threads)
__global__ void k_mlp(const float* __restrict__ colsum, float invn,
                      const float* __restrict__ w0, const float* __restrict__ b0,
                      const float* __restrict__ w1, const float* __restrict__ b1,
                      const float* __restrict__ w2, const float* __restrict__ b2,
                      float* __restrict__ out) {
  __shared__ float hg[64], y0[32], y1[16];
  int t = threadIdx.x;
  hg[t] = colsum[t] * invn;
  __syncthreads();
  if (t < 32) {
    float a = b0[t];
    for (int k = 0; k < 64; ++k) a += hg[k] * w0[k * 32 + t];
    y0[t] = fmaxf(a, 0.0f);
  }
  __syncthreads();
  if (t < 16) {
    float a = b1[t];
    for (int k = 0; k < 32; ++k) a += y0[k] * w1[k * 16 + t];
    y1[t] = fmaxf(a, 0.0f);
  }
  __syncthreads();
  if (t == 0) {
    float a = b2[0];
    for (int k = 0; k < 16; ++k) a += y1[k] * w2[k];
    out[0] = a;
  }
}

// --------------------------- host side -------------------------------------

extern "C" void kernel_launch(void* const* d_in, const int* in_sizes, int n_in,
                              void* d_out, int out_size, void* d_ws, size_t ws_size,
                              hipStream_t stream) {
  (void)n_in; (void)out_size; (void)ws_size;
  const int N = in_sizes[0];      // 50000
  const int E = in_sizes[3];      // 800000
  const int L = 4;

  const int* h_ids = (const int*)d_in[0];
  const int* e_ids = (const int*)d_in[1];
  const int* f_ids = (const int*)d_in[2];
  const int* src   = (const int*)d_in[3];
  const int* dst   = (const int*)d_in[4];
  const float* emb_h = (const float*)d_in[5];
  const float* emb_e = (const float*)d_in[6];
  const float* emb_f = (const float*)d_in[7];
  const float* A1w = (const float*)d_in[8];
  const float* A1b = (const float*)d_in[9];
  const float* A2w = (const float*)d_in[10];
  const float* A2b = (const float*)d_in[11];
  const float* B1w = (const float*)d_in[12];
  const float* B1b = (const float*)d_in[13];
  const float* B2w = (const float*)d_in[14];
  const float* B2b = (const float*)d_in[15];
  const float* B3w = (const float*)d_in[16];
  const float* B3b = (const float*)d_in[17];
  const float* C1w = (const float*)d_in[18];
  const float* C1b = (const float*)d_in[19];
  const float* C2w = (const float*)d_in[20];
  const float* C2b = (const float*)d_in[21];
  const float* bn_h_g = (const float*)d_in[22];
  const float* bn_h_b = (const float*)d_in[23];
  const float* bn_e_g = (const float*)d_in[24];
  const float* bn_e_b = (const float*)d_in[25];
  const float* mlp_w0 = (const float*)d_in[26];
  const float* mlp_b0 = (const float*)d_in[27];
  const float* mlp_w1 = (const float*)d_in[28];
  const float* mlp_b1 = (const float*)d_in[29];
  const float* mlp_w2 = (const float*)d_in[30];
  const float* mlp_b2 = (const float*)d_in[31];
  float* out = (float*)d_out;

  const int ND = N * DD;          // 3,200,000
  const int ED = E * DD;          // 51,200,000

  // workspace layout (floats)
  float* W = (float*)d_ws;
  float* h         = W;                 // ND
  float* p         = h + ND;            // ND
  float* h_new     = p + ND;            // ND
  float* p_new     = h_new + ND;        // ND
  float* B1h       = p_new + ND;        // ND
  float* B2h       = B1h + ND;          // ND
  float* vbuf      = B2h + ND;          // ND
  float* C2p       = vbuf + ND;         // ND
  float* sum_sigma = C2p + ND;          // ND
  float* stats     = sum_sigma + ND;    // 256: [h sum|sq][e sum|sq]
  float* stats_e   = stats + 128;
  float* ebuf      = stats + 256;       // ED
  float* hat       = ebuf + ED;         // ED

  const int ntiles_n = N / 16;          // 3125
  const int ntiles_e = E / 16;          // 50000
  const int WPB = 8;                    // waves per block (256 threads)
  const size_t lds_node = (size_t)7 * SLAB * sizeof(float); // 118272 B

  dim3 blk(256);
  int gN  = (ND + 255) / 256; if (gN > 8192) gN = 8192;
  int gE  = 8192;
  int gGn = (ntiles_n + WPB - 1) / WPB;
  int gGe = (ntiles_e + WPB - 1) / WPB;

  // embeddings
  k_embed_nodes<<<gN, blk, 0, stream>>>(h_ids, f_ids, emb_h, emb_f, h, p, ND);
  k_embed_edges<<<gE, blk, 0, stream>>>(e_ids, emb_e, ebuf, ED);

  for (int i = 0; i < L; ++i) {
    const float* b1w = B1w + i * DD * DD; const float* b1b = B1b + i * DD;
    const float* b2w = B2w + i * DD * DD; const float* b2b = B2b + i * DD;
    const float* b3w = B3w + i * DD * DD; const float* b3b = B3b + i * DD;
    const float* a1w = A1w + i * DD * DD; const float* a1b = A1b + i * DD;
    const float* a2w = A2w + i * 2 * DD * DD; const float* a2b = A2b + i * DD;
    const float* c1w = C1w + i * DD * DD; const float* c1b = C1b + i * DD;
    const float* c2w = C2w + i * DD * DD; const float* c2b = C2b + i * DD;

    // zero sum_sigma + stats (contiguous)
    k_zero<<<gN, blk, 0, stream>>>(sum_sigma, ND + 256);

    k_node_gemms<<<gGn, blk, lds_node, stream>>>(
        h, p, b1w, b1b, b2w, b2b, a1w, a1b, a2w, a2b, c1w, c1b, c2w, c2b,
        B1h, B2h, h_new, vbuf, p_new, C2p, ntiles_n);

    k_edge_gate<<<gGe, blk, 0, stream>>>(ebuf, b3w, b3b, B1h, B2h, src, dst,
                                         hat, sum_sigma, ntiles_e);

    k_edge_scatter<<<gE, blk, 0, stream>>>(hat, sum_sigma, vbuf, C2p, src, dst,
                                           h_new, p_new, ED);

    k_colstats<<<4096, blk, 0, stream>>>(h_new, ND, stats);
    k_colstats<<<4096, blk, 0, stream>>>(hat, ED, stats_e);

    k_finalize_node<<<gN, blk, 0, stream>>>(h, p, h_new, p_new, stats,
                                            bn_h_g + i * DD, bn_h_b + i * DD,
                                            1.0f / (float)N, ND);
    k_finalize_edge<<<gE, blk, 0, stream>>>(ebuf, hat, stats_e,
                                            bn_e_g + i * DD, bn_e_b + i * DD,
                                            1.0f / (float)E, ED);
  }

  // readout: column sums of h into stats[0:64), then tiny MLP
  k_zero<<<1, 256, 0, stream>>>(stats, 256);
  k_colstats<<<4096, blk, 0, stream>>>(h, ND, stats);
  k_mlp<<<1, 64, 0, stream>>>(stats, 1.0f / (float)N, mlp_w0, mlp_b0, mlp_w1,
                              mlp_b1, mlp_w2, mlp_b2, out);
}